// SinkhornLoss_61942018343103
// MI455X (gfx1250) — compile-verified
//
#include <hip/hip_runtime.h>
#include <hip/hip_bf16.h>
#include <math.h>
#include <stdint.h>

typedef __attribute__((ext_vector_type(2))) float v2f;
typedef __attribute__((ext_vector_type(4))) float v4f;
typedef __attribute__((ext_vector_type(8))) float v8f;
typedef __attribute__((ext_vector_type(4))) unsigned int u32x4;
typedef __attribute__((ext_vector_type(4))) int i32x4;
typedef __attribute__((ext_vector_type(8))) int i32x8;

// ---------------- constants (static shapes from the reference) ----------------
#define Bb   16
#define Nn   1024
#define Mm   1024
#define Dd   1024
#define TILE 64
#define KT   32
#define LDSTR 36   // 32 data DWORDs + 4 pad DWORDs per row (TDM pad_interval=32, pad_amount=4)

#if __has_builtin(__builtin_amdgcn_tensor_load_to_lds)
#define USE_TDM 1
#else
#define USE_TDM 0
#endif

// ---------------- squared row norms: one wave per row ----------------
__global__ __launch_bounds__(256)
void sqnorm_row(const float* __restrict__ X, float* __restrict__ out, int rows, int D) {
    int row  = blockIdx.x * (blockDim.x >> 5) + (threadIdx.x >> 5);
    int lane = threadIdx.x & 31;
    if (row >= rows) return;
    const float* xr = X + (size_t)row * D;
    float s = 0.f;
    for (int d = lane * 4; d < D; d += 128) {
        v4f v = *(const v4f*)(xr + d);
        s += v.x * v.x + v.y * v.y + v.z * v.z + v.w * v.w;
    }
    #pragma unroll
    for (int off = 16; off > 0; off >>= 1) s += __shfl_xor(s, off, 32);
    if (lane == 0) out[row] = s;
}

#if USE_TDM
// Issue one TDM 2D tile load: 64 rows x 32 f32, row stride Dd, LDS-padded to stride 36.
__device__ __forceinline__ void tdm_load_tile(const float* gptr, unsigned lds_off) {
    unsigned long long ga = (unsigned long long)(uintptr_t)gptr;
    u32x4 g0;
    g0.x = 1u;                                            // count=1, user-mode, no gather
    g0.y = lds_off;                                       // lds_addr (bytes)
    g0.z = (unsigned)(ga & 0xFFFFFFFFu);                  // global_addr[31:0]  (bits 95:64)
    g0.w = (unsigned)((ga >> 32) & 0x1FFFFFFu)            // global_addr[56:32] (bits 120:96)
         | (2u << 30);                                    // type=2 ("image")
    i32x8 g1;
    g1[0] = (2 << 16)                                     // data_size = 4 bytes
          | (1 << 20)                                     // pad_enable
          | (4 << 22)                                     // pad_interval: 2^(4+1)=32 DWORDs
          | (3 << 25);                                    // pad_amount: 3+1 = 4 DWORDs
    g1[1] = (int)((Dd & 0xFFFF) << 16);                   // abar_addr=0 | tensor_dim0[15:0]
    g1[2] = (int)(((Dd >> 16) & 0xFFFF)                   // tensor_dim0[31:16]
          | ((Nn & 0xFFFF) << 16));                       // tensor_dim1[15:0]
    g1[3] = (int)(((Nn >> 16) & 0xFFFF) | (KT << 16));    // tensor_dim1[31:16] | tile_dim0=32
    g1[4] = (int)(TILE & 0xFFFF);                         // tile_dim1=64 | tile_dim2=0 (2D)
    g1[5] = (int)Dd;                                      // tensor_dim0_stride[31:0] = 1024
    g1[6] = 0;                                            // stride hi | tensor_dim1_stride lo
    g1[7] = 0;
    i32x4 z4 = {0, 0, 0, 0};
#if __clang_major__ >= 23
    i32x8 z8 = {};
    __builtin_amdgcn_tensor_load_to_lds(g0, g1, z4, z4, z8, 0);
#else
    __builtin_amdgcn_tensor_load_to_lds(g0, g1, z4, z4, 0);
#endif
}
#endif

// ---------------- cost GEMM with f32 WMMA, TDM double-buffered LDS staging ----------------
// C[b,i,j] = 0.5*(||x_i||^2 + ||y_j||^2) - x_i . y_j ; optional transposed copy Ct.
__global__ __launch_bounds__(128)
void cost_gemm_wmma(const float* __restrict__ X, const float* __restrict__ Y,
                    const float* __restrict__ x2, const float* __restrict__ y2,
                    float* __restrict__ C, float* __restrict__ Ct) {
    __shared__ float Xs[2][TILE * LDSTR];
    __shared__ float Ys[2][TILE * LDSTR];

    const int b  = blockIdx.z;
    const int bi = blockIdx.x * TILE;     // row tile (i over N)
    const int bj = blockIdx.y * TILE;     // col tile (j over M)
    const int tid  = threadIdx.x;
    const int lane = tid & 31;
    const int wid  = tid >> 5;            // 4 waves
    const int wm   = (wid >> 1) << 5;     // wave 32x32 sub-tile origin
    const int wn   = (wid & 1) << 5;
    const int frow = lane & 15;
    const int kh   = (lane >> 4) << 1;    // K half-select for f32 16x16x4 fragments

    const float* Xb = X + (size_t)b * Nn * Dd + (size_t)bi * Dd;
    const float* Yb = Y + (size_t)b * Mm * Dd + (size_t)bj * Dd;

    v8f acc00 = {}, acc01 = {}, acc10 = {}, acc11 = {};
    const int nch = Dd / KT;

#if USE_TDM
    const unsigned offX0 = (unsigned)(uintptr_t)(&Xs[0][0]);
    const unsigned offX1 = (unsigned)(uintptr_t)(&Xs[1][0]);
    const unsigned offY0 = (unsigned)(uintptr_t)(&Ys[0][0]);
    const unsigned offY1 = (unsigned)(uintptr_t)(&Ys[1][0]);
    const bool issuer = (wid == 0);
    if (issuer) {                          // prime the pipeline: chunk 0 -> buffer 0
        tdm_load_tile(Xb, offX0);
        tdm_load_tile(Yb, offY0);
    }
    #pragma unroll 2
    for (int c = 0; c < nch; ++c) {
        const int cur = c & 1;
        if (issuer) {
            if (c + 1 < nch) {             // overlap: fetch chunk c+1 into the other buffer
                tdm_load_tile(Xb + (size_t)(c + 1) * KT, cur ? offX0 : offX1);
                tdm_load_tile(Yb + (size_t)(c + 1) * KT, cur ? offY0 : offY1);
                __builtin_amdgcn_s_wait_tensorcnt((short)2);  // chunk c landed (in-order TDM)
            } else {
                __builtin_amdgcn_s_wait_tensorcnt((short)0);
            }
        }
        __syncthreads();                   // release buffer `cur` to all waves
        const float* xs = &Xs[cur][0];
        const float* ys = &Ys[cur][0];
        #pragma unroll
        for (int kk = 0; kk < KT; kk += 4) {
            v2f a0 = *(const v2f*)&xs[(wm +      frow) * LDSTR + kk + kh];
            v2f a1 = *(const v2f*)&xs[(wm + 16 + frow) * LDSTR + kk + kh];
            v2f b0 = *(const v2f*)&ys[(wn +      frow) * LDSTR + kk + kh];
            v2f b1 = *(const v2f*)&ys[(wn + 16 + frow) * LDSTR + kk + kh];
            acc00 = __builtin_amdgcn_wmma_f32_16x16x4_f32(false, a0, false, b0, (short)0, acc00, false, false);
            acc01 = __builtin_amdgcn_wmma_f32_16x16x4_f32(false, a0, false, b1, (short)0, acc01, false, false);
            acc10 = __builtin_amdgcn_wmma_f32_16x16x4_f32(false, a1, false, b0, (short)0, acc10, false, false);
            acc11 = __builtin_amdgcn_wmma_f32_16x16x4_f32(false, a1, false, b1, (short)0, acc11, false, false);
        }
        __syncthreads();                   // reads of buffer `cur` done before TDM reuses it
    }
#else
    // Fallback: cooperative register staging
    const int lr = tid >> 3;
    const int lc = (tid & 7) << 2;
    for (int c = 0; c < nch; ++c) {
        const int cur = c & 1;
        const int k0 = c * KT;
        #pragma unroll
        for (int p = 0; p < 4; ++p) {
            int r = lr + (p << 4);
            *(v4f*)&Xs[cur][r * LDSTR + lc] = *(const v4f*)(Xb + (size_t)r * Dd + k0 + lc);
            *(v4f*)&Ys[cur][r * LDSTR + lc] = *(const v4f*)(Yb + (size_t)r * Dd + k0 + lc);
        }
        __syncthreads();
        #pragma unroll
        for (int kk = 0; kk < KT; kk += 4) {
            v2f a0 = *(const v2f*)&Xs[cur][(wm +      frow) * LDSTR + kk + kh];
            v2f a1 = *(const v2f*)&Xs[cur][(wm + 16 + frow) * LDSTR + kk + kh];
            v2f b0 = *(const v2f*)&Ys[cur][(wn +      frow) * LDSTR + kk + kh];
            v2f b1 = *(const v2f*)&Ys[cur][(wn + 16 + frow) * LDSTR + kk + kh];
            acc00 = __builtin_amdgcn_wmma_f32_16x16x4_f32(false, a0, false, b0, (short)0, acc00, false, false);
            acc01 = __builtin_amdgcn_wmma_f32_16x16x4_f32(false, a0, false, b1, (short)0, acc01, false, false);
            acc10 = __builtin_amdgcn_wmma_f32_16x16x4_f32(false, a1, false, b0, (short)0, acc10, false, false);
            acc11 = __builtin_amdgcn_wmma_f32_16x16x4_f32(false, a1, false, b1, (short)0, acc11, false, false);
        }
        __syncthreads();
    }
#endif

    // epilogue: fuse the squared-norm terms; C/D layout: VGPR r -> M = r + 8*(lane>=16), N = lane&15
    const float* x2b = x2 + (size_t)b * Nn;
    const float* y2b = y2 + (size_t)b * Mm;
    float* Cb  = C + (size_t)b * Nn * Mm;
    float* Ctb = Ct ? Ct + (size_t)b * Mm * Nn : nullptr;
    const int mo = (lane >> 4) << 3;
    const int nl = lane & 15;
    #pragma unroll
    for (int t = 0; t < 4; ++t) {
        const v8f& a = (t == 0) ? acc00 : (t == 1) ? acc01 : (t == 2) ? acc10 : acc11;
        int mt = wm + ((t >> 1) << 4);
        int nt = wn + ((t & 1) << 4);
        #pragma unroll
        for (int r = 0; r < 8; ++r) {
            int gi = bi + mt + mo + r;
            int gj = bj + nt + nl;
            float val = 0.5f * (x2b[gi] + y2b[gj]) - a[r];
            Cb[(size_t)gi * Mm + gj] = val;
            if (Ctb) Ctb[(size_t)gj * Nn + gi] = val;
        }
    }
}

// ---------------- softmin over rows: out[row] = -eps*logsumexp_j(wlog + pot[j]/eps - C[row,j]/eps) ----------------
__global__ __launch_bounds__(256)
void softmin_row(const float* __restrict__ C, const float* __restrict__ pot,
                 float* __restrict__ out, float wlog, float eps, float inv_eps,
                 int rows, int cols, int rows_per_batch) {
    int row  = blockIdx.x * (blockDim.x >> 5) + (threadIdx.x >> 5);
    int lane = threadIdx.x & 31;
    if (row >= rows) return;
    int b = row / rows_per_batch;
    const float* crow = C + (size_t)row * cols;
    float m = -INFINITY, s = 0.f;
    if (pot) {
        const float* prow = pot + (size_t)b * cols;
        for (int j = lane; j < cols; j += 32) {
            float v = wlog + prow[j] * inv_eps - crow[j] * inv_eps;
            if (v > m) { s = s * __expf(m - v) + 1.f; m = v; }
            else       { s += __expf(v - m); }
        }
    } else {
        for (int j = lane; j < cols; j += 32) {
            float v = wlog - crow[j] * inv_eps;
            if (v > m) { s = s * __expf(m - v) + 1.f; m = v; }
            else       { s += __expf(v - m); }
        }
    }
    #pragma unroll
    for (int off = 16; off > 0; off >>= 1) {
        float m2 = __shfl_xor(m, off, 32);
        float s2 = __shfl_xor(s, off, 32);
        float mm = fmaxf(m, m2);
        s = s * __expf(m - mm) + s2 * __expf(m2 - mm);
        m = mm;
    }
    if (lane == 0) out[row] = -eps * (__logf(s) + m);
}

// ---------------- averaged potential update: p = 0.5*(p + t), 4 arrays fused ----------------
__global__ __launch_bounds__(256)
void avg4(float* p0, const float* t0, float* p1, const float* t1,
          float* p2, const float* t2, float* p3, const float* t3, int n) {
    int i = blockIdx.x * blockDim.x + threadIdx.x;
    if (i < n) {
        p0[i] = 0.5f * (p0[i] + t0[i]);
        p1[i] = 0.5f * (p1[i] + t1[i]);
        p2[i] = 0.5f * (p2[i] + t2[i]);
        p3[i] = 0.5f * (p3[i] + t3[i]);
    }
}

// ---------------- final debiased divergence reduce ----------------
__global__ __launch_bounds__(256)
void loss_reduce(const float* __restrict__ fba, const float* __restrict__ faa,
                 const float* __restrict__ gab, const float* __restrict__ gbb,
                 float* __restrict__ out, int n, float scale) {
    __shared__ float red[256];
    float acc = 0.f;
    for (int i = threadIdx.x; i < n; i += 256)
        acc += (fba[i] - faa[i]) + (gab[i] - gbb[i]);
    red[threadIdx.x] = acc;
    __syncthreads();
    for (int s = 128; s > 0; s >>= 1) {
        if (threadIdx.x < s) red[threadIdx.x] += red[threadIdx.x + s];
        __syncthreads();
    }
    if (threadIdx.x == 0) out[0] = red[0] * scale;
}

// ---------------- host orchestration ----------------
extern "C" void kernel_launch(void* const* d_in, const int* in_sizes, int n_in,
                              void* d_out, int out_size, void* d_ws, size_t ws_size,
                              hipStream_t stream) {
    const float* x = (const float*)d_in[0];
    const float* y = (const float*)d_in[1];
    float* out = (float*)d_out;

    const size_t CELEMS = (size_t)Bb * Nn * Mm;          // 16M floats = 64 MB
    const int    PROWS  = Bb * Nn;                       // 16384 potentials per array

    char* w = (char*)d_ws;
    float* C_xy = (float*)w; w += CELEMS * sizeof(float);
    float* C_yx = (float*)w; w += CELEMS * sizeof(float);
    float* C_xx = (float*)w; w += CELEMS * sizeof(float);
    float* C_yy = (float*)w; w += CELEMS * sizeof(float);
    float* x2   = (float*)w; w += PROWS * sizeof(float);
    float* y2   = (float*)w; w += PROWS * sizeof(float);
    float* f_ba = (float*)w; w += PROWS * sizeof(float);
    float* g_ab = (float*)w; w += PROWS * sizeof(float);
    float* f_aa = (float*)w; w += PROWS * sizeof(float);
    float* g_bb = (float*)w; w += PROWS * sizeof(float);
    float* t_f  = (float*)w; w += PROWS * sizeof(float);
    float* t_g  = (float*)w; w += PROWS * sizeof(float);
    float* t_fs = (float*)w; w += PROWS * sizeof(float);
    float* t_gs = (float*)w; w += PROWS * sizeof(float);

    // geomloss eps schedule: [D^P] + exp(arange(P ln D, P ln blur, P ln scaling)) + [blur^P]
    double epsl[64]; int ne = 0;
    epsl[ne++] = 300.0 * 300.0;
    const double start = 2.0 * log(300.0), stop = 2.0 * log(0.3), step = 2.0 * log(0.8);
    for (double v = start; v > stop; v += step) epsl[ne++] = exp(v);
    epsl[ne++] = 0.3 * 0.3;

    const float alog = -logf((float)Nn);
    const float blog = -logf((float)Mm);

    // 1) squared norms
    sqnorm_row<<<dim3((PROWS + 7) / 8), dim3(256), 0, stream>>>(x, x2, PROWS, Dd);
    sqnorm_row<<<dim3((PROWS + 7) / 8), dim3(256), 0, stream>>>(y, y2, PROWS, Dd);

    // 2) cost matrices (f32 WMMA + TDM staging); C_yx stored so all softmins are row-sweeps
    dim3 ggrid(Nn / TILE, Mm / TILE, Bb), gblk(128);
    cost_gemm_wmma<<<ggrid, gblk, 0, stream>>>(x, y, x2, y2, C_xy, C_yx);
    cost_gemm_wmma<<<ggrid, gblk, 0, stream>>>(x, x, x2, x2, C_xx, nullptr);
    cost_gemm_wmma<<<ggrid, gblk, 0, stream>>>(y, y, y2, y2, C_yy, nullptr);

    dim3 sgrid(PROWS / 8), sblk(256);
    // 3) init at eps0 (no potentials)
    {
        float e = (float)epsl[0], ie = 1.0f / e;
        softmin_row<<<sgrid, sblk, 0, stream>>>(C_xy, nullptr, f_ba, blog, e, ie, PROWS, Mm, Nn);
        softmin_row<<<sgrid, sblk, 0, stream>>>(C_yx, nullptr, g_ab, alog, e, ie, PROWS, Nn, Mm);
        softmin_row<<<sgrid, sblk, 0, stream>>>(C_xx, nullptr, f_aa, alog, e, ie, PROWS, Nn, Nn);
        softmin_row<<<sgrid, sblk, 0, stream>>>(C_yy, nullptr, g_bb, blog, e, ie, PROWS, Mm, Mm);
    }
    // 4) epsilon-scaling symmetric Sinkhorn loop with averaged updates
    for (int it = 0; it < ne; ++it) {
        float e = (float)epsl[it], ie = 1.0f / e;
        softmin_row<<<sgrid, sblk, 0, stream>>>(C_xy, g_ab, t_f,  blog, e, ie, PROWS, Mm, Nn);
        softmin_row<<<sgrid, sblk, 0, stream>>>(C_yx, f_ba, t_g,  alog, e, ie, PROWS, Nn, Mm);
        softmin_row<<<sgrid, sblk, 0, stream>>>(C_xx, f_aa, t_fs, alog, e, ie, PROWS, Nn, Nn);
        softmin_row<<<sgrid, sblk, 0, stream>>>(C_yy, g_bb, t_gs, blog, e, ie, PROWS, Mm, Mm);
        avg4<<<dim3((PROWS + 255) / 256), dim3(256), 0, stream>>>(
            f_ba, t_f, g_ab, t_g, f_aa, t_fs, g_bb, t_gs, PROWS);
    }
    // 5) final extrapolation at last eps
    {
        float e = (float)epsl[ne - 1], ie = 1.0f / e;
        softmin_row<<<sgrid, sblk, 0, stream>>>(C_xy, g_ab, t_f,  blog, e, ie, PROWS, Mm, Nn);
        softmin_row<<<sgrid, sblk, 0, stream>>>(C_yx, f_ba, t_g,  alog, e, ie, PROWS, Nn, Mm);
        softmin_row<<<sgrid, sblk, 0, stream>>>(C_xx, f_aa, t_fs, alog, e, ie, PROWS, Nn, Nn);
        softmin_row<<<sgrid, sblk, 0, stream>>>(C_yy, g_bb, t_gs, blog, e, ie, PROWS, Mm, Mm);
    }
    // 6) loss = mean_b [ sum_i a(f_ba-f_aa) + sum_j b(g_ab-g_bb) ], a=1/N, b=1/M
    loss_reduce<<<dim3(1), dim3(256), 0, stream>>>(
        t_f, t_fs, t_g, t_gs, out, PROWS, 1.0f / (float)PROWS);
}